// GraphPropagator_21930103013404
// MI455X (gfx1250) — compile-verified
//
#include <hip/hip_runtime.h>
#include <math.h>

typedef __attribute__((ext_vector_type(2))) float v2f;
typedef __attribute__((ext_vector_type(8))) float v8f;

#define DIM 512
#define NMAX 1024
#define EEDG 16384
#define NBATCH 16
#define NODESTRIDE (NMAX * DIM)

// ---------------------------------------------------------------------------
// Generic WMMA fp32 GEMM:  Y[m][o] = act( sum_k Xrow(m,k) * W[o][k] + b[o] )
// Xrow(m,k) = k < split ? Xa[m*split+k] : Xb[m*(Din-split)+(k-split)]
// One wave (32 threads) per 16x16 output tile; K stepped by 4 via
// V_WMMA_F32_16X16X4_F32. K loop statically split over the two input
// buffers so the inner loop is pure load_b64 x2 + wmma.
// ---------------------------------------------------------------------------
__global__ void gemm_wmma_f32(const float* __restrict__ Xa,
                              const float* __restrict__ Xb,
                              int split, int Din,
                              const float* __restrict__ W,
                              const float* __restrict__ bias,
                              float* __restrict__ Y, int Dout, int doRelu,
                              long xaBatch, long xbBatch, long yBatch) {
  const int lane = threadIdx.x;
  const int half = lane >> 4;   // 0: lanes 0-15, 1: lanes 16-31
  const int lm   = lane & 15;
  const int nbase = blockIdx.x << 4;
  const int mbase = blockIdx.y << 4;
  const int b = blockIdx.z;

  const float* xa = Xa + (size_t)b * xaBatch;
  float* y = Y + (size_t)b * yBatch;

  const int m = mbase + lm;                            // A: lane indexes M
  const float* wrow = W + (size_t)(nbase + lm) * Din;  // B: lane indexes N
  const int koff = half * 2;                           // lane-half -> K pair

  v8f acc = {0.f, 0.f, 0.f, 0.f, 0.f, 0.f, 0.f, 0.f};

  // phase 1: K in [0, split) from Xa
  {
    const float* pa = xa + (size_t)m * split + koff;   // 8B aligned (koff even)
    const float* pw = wrow + koff;
    for (int k0 = 0; k0 < split; k0 += 4) {
      const v2f Af = *(const v2f*)(pa + k0);  // VGPR v -> K = koff + v
      const v2f Bf = *(const v2f*)(pw + k0);  // VGPR v -> K = koff + v
      acc = __builtin_amdgcn_wmma_f32_16x16x4_f32(false, Af, false, Bf,
                                                  (short)0, acc, false, false);
    }
  }
  // phase 2: K in [split, Din) from Xb (uniform branch)
  if (Xb) {
    const float* xb = Xb + (size_t)b * xbBatch;
    const int ldb2 = Din - split;
    const float* pb = xb + (size_t)m * ldb2 + koff;
    const float* pw = wrow + split + koff;
    for (int k0 = 0; k0 < ldb2; k0 += 4) {
      const v2f Af = *(const v2f*)(pb + k0);
      const v2f Bf = *(const v2f*)(pw + k0);
      acc = __builtin_amdgcn_wmma_f32_16x16x4_f32(false, Af, false, Bf,
                                                  (short)0, acc, false, false);
    }
  }

  const float bn = bias ? bias[nbase + lm] : 0.0f;
  for (int r = 0; r < 8; ++r) {
    const int mm = mbase + r + 8 * half;  // C/D: VGPR r -> M = r + 8*half
    float v = acc[r] + bn;
    if (doRelu) v = fmaxf(v, 0.0f);
    y[(size_t)mm * Dout + nbase + lm] = v;
  }
}

// ---------------------------------------------------------------------------
__global__ void copy_f32(const float* __restrict__ s, float* __restrict__ d,
                         long n) {
  long i = (long)blockIdx.x * blockDim.x + threadIdx.x;
  if (i < n) d[i] = s[i];
}

__global__ void zero_f32(float* p, long n) {
  long i = (long)blockIdx.x * blockDim.x + threadIdx.x;
  if (i < n) p[i] = 0.0f;
}

__global__ void init_edges(const int* __restrict__ ei, int* __restrict__ src,
                           int* __restrict__ dst, int* __restrict__ msk) {
  int t = blockIdx.x * blockDim.x + threadIdx.x;
  int b = blockIdx.y;
  if (t >= EEDG) return;
  src[b * EEDG + t] = ei[(size_t)b * 2 * EEDG + t];
  dst[b * EEDG + t] = ei[(size_t)b * 2 * EEDG + EEDG + t];
  msk[b * EEDG + t] = 1;
}

// agg[dst] += h[src] for live non-self-loop edges (agg pre-initialized to h)
__global__ void agg_edges(const float* __restrict__ H, float* __restrict__ AGG,
                          const int* __restrict__ src,
                          const int* __restrict__ dst,
                          const int* __restrict__ msk) {
  const int e = blockIdx.x;
  const int b = blockIdx.y;
  const int s = src[b * EEDG + e];
  const int d = dst[b * EEDG + e];
  if (!msk[b * EEDG + e] || s == d) return;
  const float* hs = H + (size_t)b * NODESTRIDE + (size_t)s * DIM;
  float* ad = AGG + (size_t)b * NODESTRIDE + (size_t)d * DIM;
  for (int j = threadIdx.x; j < DIM; j += blockDim.x) atomicAdd(&ad[j], hs[j]);
}

// ---------------------------------------------------------------------------
// Per-graph top-k pooling: score = x.w/||w||, bitonic sort (desc score,
// asc idx => matches jax.lax.top_k tie-breaking), gather*tanh into Xout,
// zero pad rows, emit newidx for edge relabeling. One 1024-thread block/graph.
// ---------------------------------------------------------------------------
__global__ void topk_pool(const float* __restrict__ XN,
                          const float* __restrict__ w,
                          float* __restrict__ Xout, int* __restrict__ newidx,
                          int n_in, int k, int k_pad) {
  __shared__ float wsh[DIM];
  __shared__ float ssc[NMAX];
  __shared__ int sidx[NMAX];
  __shared__ float s_wn;

  const int tid = threadIdx.x;
  const int b = blockIdx.x;
  const float* xn = XN + (size_t)b * NODESTRIDE;
  float* xo = Xout + (size_t)b * NODESTRIDE;
  int* nidx = newidx + b * NMAX;

  if (tid < DIM) wsh[tid] = w[tid];
  __syncthreads();
  if (tid == 0) {
    float s = 0.f;
    for (int j = 0; j < DIM; ++j) s += wsh[j] * wsh[j];
    s_wn = sqrtf(s);
  }
  __syncthreads();
  const float inv = 1.0f / s_wn;

  float sc = -__builtin_inff();
  if (tid < n_in) {
    const float* row = xn + (size_t)tid * DIM;
    float s = 0.f;
    for (int j = 0; j < DIM; ++j) s += row[j] * wsh[j];
    sc = s * inv;
  }
  ssc[tid] = sc;
  sidx[tid] = tid;

  // bitonic sort, "rank order" = descending score, ties by ascending index
  for (int size = 2; size <= NMAX; size <<= 1) {
    for (int stride = size >> 1; stride > 0; stride >>= 1) {
      __syncthreads();
      const int i = tid;
      const int j = i ^ stride;
      if (j > i) {
        const float si = ssc[i], sj = ssc[j];
        const int ii = sidx[i], ij = sidx[j];
        const bool iFirst = (si > sj) || (si == sj && ii < ij);
        const bool ascend = ((i & size) == 0);
        if (ascend ? !iFirst : iFirst) {
          ssc[i] = sj; ssc[j] = si; sidx[i] = ij; sidx[j] = ii;
        }
      }
    }
  }
  __syncthreads();

  if (tid < n_in) nidx[tid] = -1;
  __syncthreads();
  if (tid < k) nidx[sidx[tid]] = tid;

  // x' = x[perm] * tanh(vals); zero pad rows k..k_pad-1
  for (long t = tid; t < (long)k * DIM; t += blockDim.x) {
    const int j = (int)(t / DIM);
    const int d = (int)(t % DIM);
    xo[(size_t)j * DIM + d] = xn[(size_t)sidx[j] * DIM + d] * tanhf(ssc[j]);
  }
  for (long t = tid; t < (long)(k_pad - k) * DIM; t += blockDim.x) {
    const int j = k + (int)(t / DIM);
    const int d = (int)(t % DIM);
    xo[(size_t)j * DIM + d] = 0.0f;
  }
}

__global__ void remap_edges(int* __restrict__ src, int* __restrict__ dst,
                            int* __restrict__ msk,
                            const int* __restrict__ newidx) {
  int t = blockIdx.x * blockDim.x + threadIdx.x;
  int b = blockIdx.y;
  if (t >= EEDG) return;
  const int i = b * EEDG + t;
  const int m = msk[i];
  const int s2 = newidx[b * NMAX + src[i]];
  const int d2 = newidx[b * NMAX + dst[i]];
  const int m2 = (m && s2 >= 0 && d2 >= 0) ? 1 : 0;
  src[i] = m2 ? s2 : 0;
  dst[i] = m2 ? d2 : 0;
  msk[i] = m2;
}

// z[b][0:512] += max over rows, z[b][512:1024] += mean over rows
__global__ void readout(const float* __restrict__ Xp, float* __restrict__ z,
                        int k) {
  const int d = blockIdx.x * blockDim.x + threadIdx.x;  // 0..511
  const int b = blockIdx.y;
  const float* x = Xp + (size_t)b * NODESTRIDE;
  float mx = -__builtin_inff();
  float sum = 0.f;
  for (int j = 0; j < k; ++j) {
    const float v = x[(size_t)j * DIM + d];
    mx = fmaxf(mx, v);
    sum += v;
  }
  z[b * 1024 + d] += mx;
  z[b * 1024 + DIM + d] += sum / (float)k;
}

// ---------------------------------------------------------------------------
extern "C" void kernel_launch(void* const* d_in, const int* in_sizes, int n_in_cnt,
                              void* d_out, int out_size, void* d_ws,
                              size_t ws_size, hipStream_t stream) {
  const float* x_in    = (const float*)d_in[0];   // (16,1024,512)
  const int*   ei      = (const int*)d_in[1];     // (16,2,16384)
  const float* c_lin_W = (const float*)d_in[2];   // (3,512,512)
  const float* c_lin_b = (const float*)d_in[3];   // (3,512)
  const float* c_upd_W = (const float*)d_in[4];   // (3,512,1024)
  const float* pool_w  = (const float*)d_in[5];   // (3,512)
  const float* lin1_W  = (const float*)d_in[6];   // (512,1024)
  const float* lin1_b  = (const float*)d_in[7];   // (512,)
  const float* lin2_W  = (const float*)d_in[8];   // (256,512)
  const float* lin2_b  = (const float*)d_in[9];   // (256,)
  float* out = (float*)d_out;                     // (16,256)

  // ---- scratch carve (~138 MB) ----
  float* X   = (float*)d_ws;                 // [16][1024][512]
  float* H   = X   + (size_t)NBATCH * NODESTRIDE;
  float* AGG = H   + (size_t)NBATCH * NODESTRIDE;
  float* XN  = AGG + (size_t)NBATCH * NODESTRIDE;
  float* z   = XN  + (size_t)NBATCH * NODESTRIDE;  // [16][1024]
  float* z1  = z   + (size_t)NBATCH * 1024;        // [16][512]
  int* newidx = (int*)(z1 + (size_t)NBATCH * DIM); // [16][1024]
  int* esrc   = newidx + NBATCH * NMAX;            // [16][16384]
  int* edst   = esrc + NBATCH * EEDG;
  int* emsk   = edst + NBATCH * EEDG;

  const long nodeElems = (long)NBATCH * NODESTRIDE;

  // ---- init ----
  copy_f32<<<(int)((nodeElems + 255) / 256), 256, 0, stream>>>(x_in, X, nodeElems);
  init_edges<<<dim3((EEDG + 255) / 256, NBATCH), 256, 0, stream>>>(ei, esrc, edst, emsk);
  zero_f32<<<(NBATCH * 1024 + 255) / 256, 256, 0, stream>>>(z, NBATCH * 1024);

  const int n_in_a[3] = {1024, 820, 656};
  const int npad_a[3] = {1024, 832, 672};
  const int k_a[3]    = {820, 656, 525};
  const int kpad_a[3] = {832, 672, 528};

  for (int L = 0; L < 3; ++L) {
    const int n_in = n_in_a[L], npad = npad_a[L], k = k_a[L], kpad = kpad_a[L];

    // h = relu(X @ Wl^T + bl)
    gemm_wmma_f32<<<dim3(DIM / 16, npad / 16, NBATCH), 32, 0, stream>>>(
        X, nullptr, DIM, DIM, c_lin_W + (size_t)L * DIM * DIM,
        c_lin_b + (size_t)L * DIM, H, DIM, 1, NODESTRIDE, 0, NODESTRIDE);

    // agg = h (self loop) ...
    copy_f32<<<(int)((nodeElems + 255) / 256), 256, 0, stream>>>(H, AGG, nodeElems);
    // ... + scatter-add h[src] at dst over live non-self edges
    agg_edges<<<dim3(EEDG, NBATCH), 256, 0, stream>>>(H, AGG, esrc, edst, emsk);

    // x_new = relu(cat(agg, X) @ Wu^T)   (outer relu is idempotent)
    gemm_wmma_f32<<<dim3(DIM / 16, npad / 16, NBATCH), 32, 0, stream>>>(
        AGG, X, DIM, 2 * DIM, c_upd_W + (size_t)L * DIM * 2 * DIM, nullptr, XN,
        DIM, 1, NODESTRIDE, NODESTRIDE, NODESTRIDE);

    // top-k pool -> X (next-layer input), newidx for edge relabel
    topk_pool<<<NBATCH, 1024, 0, stream>>>(XN, pool_w + (size_t)L * DIM, X,
                                           newidx, n_in, k, kpad);
    remap_edges<<<dim3((EEDG + 255) / 256, NBATCH), 256, 0, stream>>>(
        esrc, edst, emsk, newidx);

    // z += cat(max, mean)
    readout<<<dim3(DIM / 256, NBATCH), 256, 0, stream>>>(X, z, k);
  }

  // z1 = relu(z @ lin1_W^T + lin1_b)   (16x1024 @ 1024x512)
  gemm_wmma_f32<<<dim3(DIM / 16, 1, 1), 32, 0, stream>>>(
      z, nullptr, 2 * DIM, 2 * DIM, lin1_W, lin1_b, z1, DIM, 1, 0, 0, 0);

  // out = z1 @ lin2_W^T + lin2_b       (16x512 @ 512x256)
  gemm_wmma_f32<<<dim3(256 / 16, 1, 1), 32, 0, stream>>>(
      z1, nullptr, DIM, DIM, lin2_W, lin2_b, out, 256, 0, 0, 0, 0);
}